// HartreeFockEnergy_80101140070552
// MI455X (gfx1250) — compile-verified
//
#include <hip/hip_runtime.h>

// Problem constants (match reference).
#define NPART   12
#define SIZE_   24
#define KDET    6
#define HALF    12
#define NSUB    924          // C(12,6)
#define NBASIS  800000
#define NNZ     12800000

#define BLK      256
#define GRID_B   (NBASIS / BLK)        // 3125, exact
#define GRID_C   3125
#define C_STRIDE (GRID_C * BLK)        // 800000
#define C_ITERS  (NNZ / C_STRIDE)      // 16, exact

typedef __attribute__((ext_vector_type(2))) float v2f;
typedef __attribute__((ext_vector_type(8))) float v8f;

__device__ __forceinline__ int binom_dev(int n, int k) {
  if (k < 0 || k > n) return 0;
  long long r = 1;
  for (int i = 0; i < k; ++i) r = r * (n - i) / (i + 1);
  return (int)r;
}

// Wave-wide f32 sum. Uses V_WMMA_F32_16X16X4_F32 with A=ones(16x4),
// B(4x16) holding the 32 per-lane partials (other slot zero): D columns are
// column sums, i.e. lane l ends up holding p_{l&15} + p_{(l&15)+16}.
// Four shfl_xor steps finish the 16-group. Requires EXEC == all-ones
// (guaranteed: callers use exact grid sizes with no divergence here).
__device__ __forceinline__ float wave_sum32(float p) {
  v2f a; a[0] = 1.0f; a[1] = 1.0f;   // A = all ones
  v2f b; b[0] = p;    b[1] = 0.0f;   // one partial per lane
  v8f c = {};
  c = __builtin_amdgcn_wmma_f32_16x16x4_f32(false, a, false, b,
                                            (short)0, c, false, false);
  float s = c[0];
  s += __shfl_xor(s, 8, 32);
  s += __shfl_xor(s, 4, 32);
  s += __shfl_xor(s, 2, 32);
  s += __shfl_xor(s, 1, 32);
  return s;
}

// ---------------------------------------------------------------------------
// Kernel A: precompute all 924 subset determinants for each spin sector.
// Unrank r -> sorted 6-subset of [0,12) via combinadics, then pivoted
// complex 6x6 LU. Runs once over 924 threads; cost is negligible.
// ---------------------------------------------------------------------------
__global__ void det_table_kernel(const float* __restrict__ w,
                                 float2* __restrict__ detA,
                                 float2* __restrict__ detB) {
  int r = blockIdx.x * blockDim.x + threadIdx.x;
  if (r >= NSUB) return;

  int cols[KDET];
  int rem = r;
  for (int k = KDET; k >= 1; --k) {           // combinadic unranking
    int c = k - 1;
    for (int t = k - 1; t < HALF; ++t)
      if (binom_dev(t, k) <= rem) c = t;
    cols[k - 1] = c;
    rem -= binom_dev(c, k);
  }

  for (int spin = 0; spin < 2; ++spin) {
    int row0 = spin * KDET;
    float re[KDET][KDET], im[KDET][KDET];
    for (int p = 0; p < KDET; ++p)
      for (int j = 0; j < KDET; ++j) {
        int s = cols[j];
        re[p][j] = w[(row0 + p) * SIZE_ + s];                 // weights[0]
        im[p][j] = w[NPART * SIZE_ + (row0 + p) * SIZE_ + s]; // weights[1]
      }

    float dre = 1.f, dim_ = 0.f;
    for (int k = 0; k < KDET; ++k) {
      int piv = k;
      float best = fabsf(re[k][k]) + fabsf(im[k][k]);
      for (int i = k + 1; i < KDET; ++i) {
        float m = fabsf(re[i][k]) + fabsf(im[i][k]);
        if (m > best) { best = m; piv = i; }
      }
      if (piv != k) {
        for (int j = 0; j < KDET; ++j) {
          float tr = re[k][j]; re[k][j] = re[piv][j]; re[piv][j] = tr;
          float ti = im[k][j]; im[k][j] = im[piv][j]; im[piv][j] = ti;
        }
        dre = -dre; dim_ = -dim_;
      }
      float ar = re[k][k], ai = im[k][k];
      float nr = dre * ar - dim_ * ai;
      float ni = dre * ai + dim_ * ar;
      dre = nr; dim_ = ni;
      float inv = 1.0f / (ar * ar + ai * ai);
      float cr = ar * inv, ci = -ai * inv;          // 1 / pivot
      for (int i = k + 1; i < KDET; ++i) {
        float lr = re[i][k] * cr - im[i][k] * ci;
        float li = re[i][k] * ci + im[i][k] * cr;
        for (int j = k + 1; j < KDET; ++j) {
          re[i][j] -= lr * re[k][j] - li * im[k][j];
          im[i][j] -= lr * im[k][j] + li * re[k][j];
        }
      }
    }
    float2 d = make_float2(dre, dim_);
    if (spin == 0) detA[r] = d; else detB[r] = d;
  }
}

// ---------------------------------------------------------------------------
// Kernel B: per-state amplitude via table lookup + block partial of sum(psi^2).
// Streams 38.4 MB of indices, tables live in LDS.
// ---------------------------------------------------------------------------
__global__ void __launch_bounds__(BLK)
psi_kernel(const int* __restrict__ idx_a, const int* __restrict__ idx_b,
           const float2* __restrict__ detA, const float2* __restrict__ detB,
           float* __restrict__ psi, float* __restrict__ den_partial) {
  __shared__ float2 sA[NSUB];
  __shared__ float2 sB[NSUB];
  __shared__ int    sBin[13 * 7];
  __shared__ float  wsum[BLK / 32];

  int tid = threadIdx.x;
  for (int i = tid; i < NSUB; i += BLK) { sA[i] = detA[i]; sB[i] = detB[i]; }
  for (int i = tid; i < 13 * 7; i += BLK) sBin[i] = binom_dev(i / 7, i % 7);
  __syncthreads();

  int n = blockIdx.x * BLK + tid;                 // exact grid: always valid
  const int* ia = idx_a + n * 6;
  const int* ib = idx_b + n * 6;
  int ra = sBin[ia[0] * 7 + 1] + sBin[ia[1] * 7 + 2] + sBin[ia[2] * 7 + 3]
         + sBin[ia[3] * 7 + 4] + sBin[ia[4] * 7 + 5] + sBin[ia[5] * 7 + 6];
  int rb = sBin[ib[0] * 7 + 1] + sBin[ib[1] * 7 + 2] + sBin[ib[2] * 7 + 3]
         + sBin[ib[3] * 7 + 4] + sBin[ib[4] * 7 + 5] + sBin[ib[5] * 7 + 6];

  float2 da = sA[ra];
  float2 db = sB[rb];
  float p = da.x * db.x - da.y * db.y;            // Re(detA * detB)
  psi[n] = p;

  float s = wave_sum32(p * p);
  if ((tid & 31) == 0) wsum[tid >> 5] = s;
  __syncthreads();
  if (tid == 0) {
    float t = 0.f;
    for (int i = 0; i < BLK / 32; ++i) t += wsum[i];
    den_partial[blockIdx.x] = t;
  }
}

// ---------------------------------------------------------------------------
// Kernel C: COO contraction  sum(v * psi[row] * psi[col]).
// Streams 153.6 MB; psi (3.2 MB) is L2-resident so gathers hit L2.
// ---------------------------------------------------------------------------
__global__ void __launch_bounds__(BLK)
coo_kernel(const int* __restrict__ rows, const int* __restrict__ colsv,
           const float* __restrict__ vals, const float* __restrict__ psi,
           float* __restrict__ num_partial) {
  __shared__ float wsum[BLK / 32];
  int tid = threadIdx.x;
  long long base = (long long)blockIdx.x * BLK + tid;

  float acc = 0.f;
#pragma unroll
  for (int it = 0; it < C_ITERS; ++it) {          // exact: no tail guard
    long long i = base + (long long)it * C_STRIDE;
    __builtin_prefetch(vals + i + 4ll * C_STRIDE, 0, 0);  // global_prefetch_b8
    int r = rows[i];
    int c = colsv[i];
    float v = vals[i];
    acc += v * psi[r] * psi[c];
  }

  float s = wave_sum32(acc);
  if ((tid & 31) == 0) wsum[tid >> 5] = s;
  __syncthreads();
  if (tid == 0) {
    float t = 0.f;
    for (int i = 0; i < BLK / 32; ++i) t += wsum[i];
    num_partial[blockIdx.x] = t;
  }
}

// ---------------------------------------------------------------------------
// Kernel D: deterministic final combine, E = num / den.
// ---------------------------------------------------------------------------
__global__ void __launch_bounds__(BLK)
final_kernel(const float* __restrict__ den_partial,
             const float* __restrict__ num_partial,
             int nden, int nnum, float* __restrict__ out) {
  __shared__ float sh[2 * (BLK / 32)];
  int tid = threadIdx.x;
  float d = 0.f, nm = 0.f;
  for (int i = tid; i < nden; i += BLK) d  += den_partial[i];
  for (int i = tid; i < nnum; i += BLK) nm += num_partial[i];
  float sd = wave_sum32(d);
  float sn = wave_sum32(nm);
  if ((tid & 31) == 0) { sh[tid >> 5] = sd; sh[(BLK / 32) + (tid >> 5)] = sn; }
  __syncthreads();
  if (tid == 0) {
    float D = 0.f, N = 0.f;
    for (int i = 0; i < BLK / 32; ++i) { D += sh[i]; N += sh[BLK / 32 + i]; }
    out[0] = N / D;
  }
}

extern "C" void kernel_launch(void* const* d_in, const int* in_sizes, int n_in,
                              void* d_out, int out_size, void* d_ws, size_t ws_size,
                              hipStream_t stream) {
  const float* weights = (const float*)d_in[0];
  const int*   idx_a   = (const int*)d_in[1];
  const int*   idx_b   = (const int*)d_in[2];
  const int*   rows    = (const int*)d_in[3];
  const int*   colsv   = (const int*)d_in[4];
  const float* vals    = (const float*)d_in[5];
  float* out = (float*)d_out;

  // Workspace layout (all offsets 16B-aligned): det tables, psi, partials.
  char* ws = (char*)d_ws;
  float2* detA        = (float2*)(ws);
  float2* detB        = (float2*)(ws + 8192);
  float*  psi         = (float*)(ws + 16384);
  float*  den_partial = (float*)(ws + 16384 + (size_t)NBASIS * 4);
  float*  num_partial = (float*)(ws + 16384 + (size_t)NBASIS * 4 + 16384);

  det_table_kernel<<<(NSUB + BLK - 1) / BLK, BLK, 0, stream>>>(weights, detA, detB);
  psi_kernel<<<GRID_B, BLK, 0, stream>>>(idx_a, idx_b, detA, detB, psi, den_partial);
  coo_kernel<<<GRID_C, BLK, 0, stream>>>(rows, colsv, vals, psi, num_partial);
  final_kernel<<<1, BLK, 0, stream>>>(den_partial, num_partial, GRID_B, GRID_C, out);
}